// LandmarkLoss_420906795568
// MI455X (gfx1250) — compile-verified
//
#include <hip/hip_runtime.h>

// ---------------------------------------------------------------------------
// LandmarkLoss on MI455X (gfx1250, wave32)
//
// ||c - t||^2 = ||c||^2 + ||t||^2 - 2 c.t : the N x K distance matrices are
// (Nx3)x(3xK) GEMMs on V_WMMA_F32_16X16X4_F32 (16 points x 16 targets,
// inner dim = x,y,z,0). One wave = 16 points vs 64 targets; the A-tiles,
// ||t||^2 (raw + masked) and the per-target g2p running min all live in
// per-lane REGISTERS (lane's target rows are point-independent), so the hot
// loop has zero LDS reads and only 5 ds_bpermute ops (top-2 half merge).
// All per-element logic is branchless (v_cndmask, no exec saves).
// ---------------------------------------------------------------------------

typedef float v2f __attribute__((ext_vector_type(2)));
typedef float v8f __attribute__((ext_vector_type(8)));

#define DIST_THRESH 0.1f
#define W_DIST      0.05f
#define W_CHAMFER   0.05f
#define W_SEP       0.0005f
#define KT          64
#define BATCHES     4

// Branchless top-2 (key dc, payload lc) -> all selects, no branches.
__device__ __forceinline__ void top2_ins(float dc, float lc,
                                         float& d0, float& l0,
                                         float& d1, float& l1) {
  const bool c0 = dc < d0;
  const bool c1 = dc < d1;
  const float nd1 = c0 ? d0 : (c1 ? dc : d1);
  const float nl1 = c0 ? l0 : (c1 ? lc : l1);
  d1 = nd1; l1 = nl1;
  l0 = c0 ? lc : l0;
  d0 = c0 ? dc : d0;
}

__global__ void landmark_init(float* accum, unsigned* gmin) {
  int i = blockIdx.x * blockDim.x + threadIdx.x;
  if (i < BATCHES * 3)  accum[i] = 0.0f;
  if (i < BATCHES * KT) gmin[i]  = 0x7F800000u;  // +inf bits
}

__global__ void __launch_bounds__(256) landmark_main(
    const float* __restrict__ C, const float4* __restrict__ F4,
    const float* __restrict__ tC, const int* __restrict__ tF,
    const int* __restrict__ cls, int nCls, int N,
    float* __restrict__ accum, unsigned* __restrict__ gmin) {
  __shared__ float    s_tx[KT], s_ty[KT], s_tz[KT], s_tn2[KT], s_minf[KT];
  __shared__ unsigned s_gmin[KT];
  __shared__ float    s_wsum[8][3];

  const int   b   = blockIdx.y;
  const int   tid = threadIdx.x;
  const float INF = __builtin_inff();

  if (tid < KT) {
    float tx = tC[(b * KT + tid) * 3 + 0];
    float ty = tC[(b * KT + tid) * 3 + 1];
    float tz = tC[(b * KT + tid) * 3 + 2];
    int   c  = tF[b * KT + tid];
    bool valid = false;
    for (int j = 0; j < nCls; ++j) valid |= (c == cls[j]);
    s_tx[tid] = tx; s_ty[tid] = ty; s_tz[tid] = tz;
    s_tn2[tid]  = tx * tx + ty * ty + tz * tz;
    s_minf[tid] = valid ? 0.0f : INF;   // additive mask: +inf kills invalid k
    s_gmin[tid] = 0x7F800000u;
  }
  __syncthreads();

  const int w    = tid >> 5;
  const int lane = tid & 31;
  const int half = lane >> 4;   // 0: carries K=0,1 (x,y) ; 1: K=2,3 (z,0)
  const int p    = lane & 15;   // point (B column) and target (A row) index

  // ---- per-lane loop-invariant registers -------------------------------
  v2f   aTr[4];                 // A-tiles: 16 targets x (x,y | z,0)
  float tn2r[32], tn2mr[32];    // ||t||^2 and masked ||t||^2 for lane's rows
  float gminr[32];              // running min ld^2 per lane's target row
#pragma unroll
  for (int tt = 0; tt < 4; ++tt) {
    const int mA = tt * 16 + p;
    aTr[tt].x = half ? s_tz[mA] : s_tx[mA];
    aTr[tt].y = half ? 0.0f     : s_ty[mA];
#pragma unroll
    for (int e = 0; e < 8; ++e) {
      const int i = tt * 8 + e;
      const int m = tt * 16 + e + (half << 3);
      tn2r[i]  = s_tn2[m];
      tn2mr[i] = s_tn2[m] + s_minf[m];
      gminr[i] = INF;
    }
  }

  float acc_h = 0.0f, acc_c = 0.0f, acc_s = 0.0f;

  const int numTiles = (N + 15) >> 4;
  for (int t = blockIdx.x * 8 + w; t < numTiles; t += gridDim.x * 8) {
    const int  n0     = t * 16 + p;
    const bool pvalid = (n0 < N);
    const int  n      = pvalid ? n0 : (N - 1);   // clamp: always load in-bounds
    const long base   = (long)b * N + n;

    const float  x = C[base * 3 + 0];
    const float  y = C[base * 3 + 1];
    const float  z = C[base * 3 + 2];
    const float4 f = F4[base];                   // 16B aligned -> b128
    const float qx = x + f.y, qy = y + f.z, qz = z + f.w;

    const float pinf = pvalid ? 0.0f : INF;      // poisons invalid points
    const float pn2  = x * x + y * y + z * z + pinf;
    const float qn2  = qx * qx + qy * qy + qz * qz + pinf;

    // B matrices (4x16): lanes 0-15 -> K=0,1 ; lanes 16-31 -> K=2,3
    v2f bC, bQ;
    bC.x = half ? z  : x;  bC.y = half ? 0.0f : y;
    bQ.x = half ? qz : qx; bQ.y = half ? 0.0f : qy;

    float d0 = INF, l0 = INF, d1 = INF, l1 = INF; // top-2 masked pd^2 / ld^2 payload
    float ldm2 = INF;                             // masked min ld^2 (p2g)

#pragma unroll
    for (int tt = 0; tt < 4; ++tt) {
      v8f zero8 = {};
      v8f dC = __builtin_amdgcn_wmma_f32_16x16x4_f32(
          false, aTr[tt], false, bC, (short)0, zero8, false, false);
      v8f dQ = __builtin_amdgcn_wmma_f32_16x16x4_f32(
          false, aTr[tt], false, bQ, (short)0, zero8, false, false);
#pragma unroll
      for (int e = 0; e < 8; ++e) {
        const int i = tt * 8 + e;
        const float pd2m = fmaxf(tn2mr[i] + pn2 - 2.0f * dC[e], 0.0f);
        const float ld2  = fmaxf(tn2r[i]  + qn2 - 2.0f * dQ[e], 0.0f);
        const float ld2m = fmaxf(tn2mr[i] + qn2 - 2.0f * dQ[e], 0.0f);
        top2_ins(pd2m, ld2, d0, l0, d1, l1);
        ldm2     = fminf(ldm2, ld2m);
        gminr[i] = fminf(gminr[i], ld2);          // g2p: deferred reduction
      }
    }

    // merge the two halves (each saw 32 of the 64 targets for point p)
    const float d0p = __shfl_xor(d0, 16, 32), l0p = __shfl_xor(l0, 16, 32);
    const float d1p = __shfl_xor(d1, 16, 32), l1p = __shfl_xor(l1, 16, 32);
    top2_ins(d0p, l0p, d0, l0, d1, l1);
    top2_ins(d1p, l1p, d0, l0, d1, l1);
    ldm2 = fminf(ldm2, __shfl_xor(ldm2, 16, 32));

    // branchless epilogue (only lanes 0-15 with valid points contribute)
    const bool  active = (half == 0) && pvalid;
    const float min_d  = sqrtf(d0);
    const bool  pm     = (min_d < DIST_THRESH);
    const float tgt    = fminf(min_d, 2.0f * DIST_THRESH);
    const float diff   = f.x - tgt;
    const float ad     = fabsf(diff);
    const float hub    = (ad < 1.0f) ? (0.5f * diff * diff) : (ad - 0.5f);
    acc_h += active ? hub : 0.0f;
    acc_c += (active && pm) ? ldm2 : 0.0f;
    const float ratio = sqrtf(l0) / sqrtf(l1);
    acc_s += (active && pm) ? ratio : 0.0f;
  }

  // ---- deferred g2p: cross-lane min once per wave, then LDS atomic ------
#pragma unroll
  for (int i = 0; i < 32; ++i) {
    float g = gminr[i];
    g = fminf(g, __shfl_xor(g, 1, 32));
    g = fminf(g, __shfl_xor(g, 2, 32));
    g = fminf(g, __shfl_xor(g, 4, 32));
    g = fminf(g, __shfl_xor(g, 8, 32));
    if (p == 0 && g < INF) {
      const int m = (i >> 3) * 16 + (i & 7) + (half << 3);
      atomicMin(&s_gmin[m], __float_as_uint(g));
    }
  }

  // ---- block reduction of the three scalar sums -------------------------
  for (int s = 16; s > 0; s >>= 1) {
    acc_h += __shfl_xor(acc_h, s, 32);
    acc_c += __shfl_xor(acc_c, s, 32);
    acc_s += __shfl_xor(acc_s, s, 32);
  }
  if (lane == 0) { s_wsum[w][0] = acc_h; s_wsum[w][1] = acc_c; s_wsum[w][2] = acc_s; }
  __syncthreads();
  if (tid == 0) {
    float h = 0.f, c2 = 0.f, s2 = 0.f;
    for (int i = 0; i < 8; ++i) { h += s_wsum[i][0]; c2 += s_wsum[i][1]; s2 += s_wsum[i][2]; }
    atomicAdd(&accum[b * 3 + 0], h);
    atomicAdd(&accum[b * 3 + 1], c2);
    atomicAdd(&accum[b * 3 + 2], s2);
  }
  if (tid < KT) atomicMin(&gmin[b * KT + tid], s_gmin[tid]);
}

__global__ void __launch_bounds__(256) landmark_final(
    const float* __restrict__ accum, const unsigned* __restrict__ gmin,
    const int* __restrict__ tF, const int* __restrict__ cls, int nCls,
    float* __restrict__ out) {
  __shared__ float s_g[256];
  __shared__ int   s_nv[256];
  __shared__ float s_loss[BATCHES];
  const int tid = threadIdx.x;
  const int b = tid >> 6, k = tid & 63;

  const int c = tF[b * KT + k];
  bool valid = false;
  for (int j = 0; j < nCls; ++j) valid |= (c == cls[j]);
  const float g2 = __uint_as_float(gmin[b * KT + k]);
  s_g[tid]  = (valid && g2 < __builtin_inff()) ? g2 : 0.0f;  // g2p^2 = min ld^2
  s_nv[tid] = valid ? 1 : 0;
  __syncthreads();
  for (int off = 32; off > 0; off >>= 1) {
    if (k < off) { s_g[tid] += s_g[tid + off]; s_nv[tid] += s_nv[tid + off]; }
    __syncthreads();
  }
  if (k == 0) {
    const float hub  = accum[b * 3 + 0];
    const float p2g2 = accum[b * 3 + 1];
    const float sep  = accum[b * 3 + 2];
    const float chamfer = p2g2 + s_g[tid];
    const float sepg    = (s_nv[tid] >= 2) ? sep : 0.0f;
    s_loss[b] = W_DIST * hub + W_CHAMFER * chamfer + W_SEP * sepg;
  }
  __syncthreads();
  if (tid == 0)
    out[0] = 0.25f * (s_loss[0] + s_loss[1] + s_loss[2] + s_loss[3]);
}

extern "C" void kernel_launch(void* const* d_in, const int* in_sizes, int n_in,
                              void* d_out, int out_size, void* d_ws, size_t ws_size,
                              hipStream_t stream) {
  const float*  C   = (const float*)d_in[0];   // (B,N,3)
  const float4* F4  = (const float4*)d_in[1];  // (B,N,4) -> float4 rows
  const float*  tC  = (const float*)d_in[2];   // (B,K,3)
  const int*    tF  = (const int*)d_in[3];     // (B,K)
  const int*    cls = (const int*)d_in[4];     // (6,)
  const int nCls = in_sizes[4];
  const int N    = in_sizes[0] / (3 * BATCHES);

  float*    accum = (float*)d_ws;                   // BATCHES*3 floats
  unsigned* gmin  = (unsigned*)((char*)d_ws + 64);  // BATCHES*KT uints

  landmark_init<<<1, 512, 0, stream>>>(accum, gmin);

  const int numTiles = (N + 15) / 16;
  int bx = (numTiles + 7) / 8;
  if (bx > 2048) bx = 2048;  // grid-stride covers the rest
  dim3 grid(bx, BATCHES);
  landmark_main<<<grid, 256, 0, stream>>>(C, F4, tC, tF, cls, nCls, N, accum, gmin);

  landmark_final<<<1, 256, 0, stream>>>(accum, gmin, tF, cls, nCls, (float*)d_out);
}